// GraphSAGE_73332271612561
// MI455X (gfx1250) — compile-verified
//
#include <hip/hip_runtime.h>
#include <math.h>

// ---------------------------------------------------------------------------
// GraphSAGE-style 3-layer GCN for MI455X (gfx1250, wave32, WMMA + TDM).
//   layer: agg = scatter_add(h[src] -> dst);  h = relu(agg @ W + b)
//   final: logits = agg @ W2 + b2; out = log_softmax(logits)
// ---------------------------------------------------------------------------

#define N_NODES 50000
#define N_EDGES 800000
#define FEAT    128
#define HID     128
#define CLASSES 40

typedef __attribute__((ext_vector_type(2))) float v2f;
typedef __attribute__((ext_vector_type(8))) float v8f;
typedef unsigned int u32x4 __attribute__((ext_vector_type(4)));
typedef int          i32x4 __attribute__((ext_vector_type(4)));
typedef int          i32x8 __attribute__((ext_vector_type(8)));

#ifndef __has_builtin
#define __has_builtin(x) 0
#endif
#if __has_builtin(__builtin_amdgcn_tensor_load_to_lds) && \
    __has_builtin(__builtin_amdgcn_s_wait_tensorcnt)
#define HAVE_TDM 1
#else
#define HAVE_TDM 0
#endif

// ---------------------------------------------------------------------------
// Zero a float buffer (float4 granularity). Graph-capture-safe.
// ---------------------------------------------------------------------------
__global__ void zero_kernel(float4* __restrict__ p, int n4) {
    int i = blockIdx.x * blockDim.x + threadIdx.x;
    if (i < n4) p[i] = make_float4(0.f, 0.f, 0.f, 0.f);
}

// ---------------------------------------------------------------------------
// Edge aggregation: one wave (32 lanes) per edge. Each lane moves one float4
// (32 lanes x 16B = 512B = one 128-float feature row), gathered coalesced
// from h[src] and scatter-added into agg[dst] with f32 atomics. h and agg are
// both 25.6MB -> L2-resident (192MB), so atomics resolve at L2 throughput.
// ---------------------------------------------------------------------------
__global__ __launch_bounds__(256)
void aggregate_kernel(const float* __restrict__ h,
                      const int*   __restrict__ src,
                      const int*   __restrict__ dst,
                      float*       __restrict__ agg) {
    const int lane = threadIdx.x & 31;
    const int wave = threadIdx.x >> 5;
    const int e = blockIdx.x * (blockDim.x >> 5) + wave;
    if (e >= N_EDGES) return;

    const int s = src[e];
    const int d = dst[e];

    const float4 v = ((const float4*)(h + (size_t)s * FEAT))[lane];
    float* out = agg + (size_t)d * FEAT + lane * 4;
    atomicAdd(out + 0, v.x);
    atomicAdd(out + 1, v.y);
    atomicAdd(out + 2, v.z);
    atomicAdd(out + 3, v.w);
}

// ---------------------------------------------------------------------------
// Dense GEMM via V_WMMA_F32_16X16X4_F32.
//   A: [N_NODES x 128]; one 16-row strip per block, staged to LDS by the
//      Tensor Data Mover (TDM). pad_interval=6 (128 DW) + pad_amount=0 (1 DW)
//      produce a 129-float row stride -> conflict-free LDS reads
//      (bank = (m + k) % 64).
//   W: [128 x NCOLS], read through L1/L2 (shared by all 3125 blocks).
//   One wave per 16-col output tile; K-loop of 32 WMMAs, branch-free:
//   out-of-range columns are address-clamped (they only feed D columns that
//   the guarded store discards).
// Fragment layouts per CDNA5 ISA 7.12.2:
//   A 16x4 f32 : lane m=lane%16 holds M=m; vgpr{0,1} = K = (lane/16)*2 + {0,1}
//   B 4x16 f32 : lane n=lane%16 holds N=n; vgpr{0,1} = K = (lane/16)*2 + {0,1}
//   D 16x16 f32: vgpr r -> row r + (lane/16)*8, col = lane%16
// ---------------------------------------------------------------------------
template<int NCOLS, bool RELU>
__global__ void gemm_wmma_kernel(const float* __restrict__ A,
                                 const float* __restrict__ W,
                                 const float* __restrict__ bias,
                                 float*       __restrict__ out) {
    constexpr int K = 128;
    __shared__ float As[16 * 129];

    const int row0 = blockIdx.x * 16;   // N_NODES = 3125 * 16, no row guard

#if HAVE_TDM
    if (threadIdx.x < 32) {
        // Tensor DMA descriptor (ISA ch.8): 2D tile, 16 rows x 128 f32.
        const unsigned long long ga =
            (unsigned long long)(const void*)(A + (size_t)row0 * K);
        const unsigned lds = (unsigned)(unsigned long long)(const void*)&As[0];

        u32x4 g0;
        g0[0] = 1u;                                      // count=1, user desc
        g0[1] = lds;                                     // lds_addr (bytes)
        g0[2] = (unsigned)(ga & 0xFFFFFFFFu);            // global_addr[31:0]
        g0[3] = (unsigned)((ga >> 32) & 0x01FFFFFFull)   // global_addr[56:32]
              | (2u << 30);                              // type=2 ("image")

        i32x8 g1;
        g1[0] = (2 << 16)        // data_size = 4 bytes
              | (1 << 20)        // pad_enable
              | (6 << 22);       // pad_interval: 128 DWORDs
                                 // pad_amount = 0 -> 1 DWORD pad
        g1[1] = (int)(128u << 16);  // tensor_dim0[15:0] = 128
        g1[2] = (int)(16u  << 16);  // tensor_dim0 hi=0 | tensor_dim1[15:0]=16
        g1[3] = (int)(128u << 16);  // tensor_dim1 hi=0 | tile_dim0=128
        g1[4] = 16;                 // tile_dim1=16, tile_dim2=0
        g1[5] = 128;                // tensor_dim0_stride[31:0] = 128
        g1[6] = 0;                  // stride hi / tensor_dim1_stride lo
        g1[7] = 0;

        const i32x4 gz4 = {0, 0, 0, 0};
        const i32x8 gz8 = {0, 0, 0, 0, 0, 0, 0, 0};
        // clang-23 / therock-10.0 6-arg form:
        // (g0, g1, g2, g3, extra, cpol)
        __builtin_amdgcn_tensor_load_to_lds(g0, g1, gz4, gz4, gz8, 0);
        __builtin_amdgcn_s_wait_tensorcnt(0);
    }
    __syncthreads();
#else
    for (int idx = threadIdx.x; idx < 16 * K; idx += blockDim.x) {
        const int r = idx >> 7;
        const int c = idx & 127;
        As[r * 129 + c] = A[(size_t)(row0 + r) * K + c];
    }
    __syncthreads();
#endif

    const int lane = threadIdx.x & 31;
    const int wv   = threadIdx.x >> 5;      // column tile index
    const int n    = lane & 15;             // M (for A) / N (for B,D)
    const int hi   = lane >> 4;             // K-pair selector
    const int col0 = wv * 16;
    const int col  = col0 + n;

    // Clamped column: keeps loads in-bounds and the k-loop branch-free.
    const int colc = min(col, NCOLS - 1);
    const float bcol = bias[colc];
    const float* __restrict__ Wp = W + colc;

    v8f acc = {};
    #pragma unroll
    for (int k0 = 0; k0 < K; k0 += 4) {
        const int kk = k0 + hi * 2;
        v2f a, b;
        a.x = As[n * 129 + kk];
        a.y = As[n * 129 + kk + 1];
        b.x = Wp[(size_t)kk * NCOLS];
        b.y = Wp[(size_t)(kk + 1) * NCOLS];
        // (neg_a, A, neg_b, B, c_mod, C, reuse_a, reuse_b)
        acc = __builtin_amdgcn_wmma_f32_16x16x4_f32(
            false, a, false, b, (short)0, acc, false, false);
    }

    if (col < NCOLS) {
        #pragma unroll
        for (int r = 0; r < 8; ++r) {
            const int row = row0 + r + hi * 8;
            float v = acc[r] + bcol;
            if (RELU) v = fmaxf(v, 0.f);
            out[(size_t)row * NCOLS + col] = v;
        }
    }
}

// ---------------------------------------------------------------------------
// In-place log_softmax over CLASSES=40 logits; one wave per node.
// Lane covers class c = lane, and c = lane+32 for lanes 0..7.
// ---------------------------------------------------------------------------
__global__ __launch_bounds__(256)
void log_softmax_kernel(float* __restrict__ out) {
    const int lane = threadIdx.x & 31;
    const int wave = threadIdx.x >> 5;
    const int node = blockIdx.x * (blockDim.x >> 5) + wave;
    if (node >= N_NODES) return;

    float* row = out + (size_t)node * CLASSES;
    const bool has2 = (lane + 32) < CLASSES;

    const float v1 = row[lane];
    const float v2 = has2 ? row[lane + 32] : -INFINITY;

    float m = fmaxf(v1, v2);
    #pragma unroll
    for (int off = 16; off >= 1; off >>= 1)
        m = fmaxf(m, __shfl_xor(m, off, 32));

    float s = expf(v1 - m) + (has2 ? expf(v2 - m) : 0.f);
    #pragma unroll
    for (int off = 16; off >= 1; off >>= 1)
        s += __shfl_xor(s, off, 32);

    const float lse = m + logf(s);
    row[lane] = v1 - lse;
    if (has2) row[lane + 32] = v2 - lse;
}

// ---------------------------------------------------------------------------
// Launcher. Inputs in setup_inputs() order:
//   0:x 1:src 2:dst 3:W1 4:b1 5:Wm 6:bm 7:W2 8:b2
// Workspace layout: [agg: 50000*128 f32][hbuf: 50000*128 f32]  (51.2 MB)
// ---------------------------------------------------------------------------
extern "C" void kernel_launch(void* const* d_in, const int* in_sizes, int n_in,
                              void* d_out, int out_size, void* d_ws, size_t ws_size,
                              hipStream_t stream) {
    const float* x   = (const float*)d_in[0];
    const int*   src = (const int*)d_in[1];
    const int*   dst = (const int*)d_in[2];
    const float* W1  = (const float*)d_in[3];
    const float* b1  = (const float*)d_in[4];
    const float* Wm  = (const float*)d_in[5];
    const float* bm  = (const float*)d_in[6];
    const float* W2  = (const float*)d_in[7];
    const float* b2  = (const float*)d_in[8];
    float* out = (float*)d_out;

    float* agg  = (float*)d_ws;
    float* hbuf = agg + (size_t)N_NODES * HID;

    const int n4 = N_NODES * HID / 4;                 // 1.6M float4
    const dim3 zgrid((n4 + 255) / 256);
    const dim3 agrid(N_EDGES / 8);                    // 8 edges (waves) / block
    const dim3 ggrid(N_NODES / 16);                   // 3125 row strips
    const dim3 sgrid((N_NODES + 7) / 8);

    // Layer 1: agg = scatter(x); h1 = relu(agg @ W1 + b1)
    zero_kernel<<<zgrid, 256, 0, stream>>>((float4*)agg, n4);
    aggregate_kernel<<<agrid, 256, 0, stream>>>(x, src, dst, agg);
    gemm_wmma_kernel<HID, true><<<ggrid, 8 * 32, 0, stream>>>(agg, W1, b1, hbuf);

    // Layer 2: agg = scatter(h1); h2 = relu(agg @ Wm + bm)   (hbuf reused)
    zero_kernel<<<zgrid, 256, 0, stream>>>((float4*)agg, n4);
    aggregate_kernel<<<agrid, 256, 0, stream>>>(hbuf, src, dst, agg);
    gemm_wmma_kernel<HID, true><<<ggrid, 8 * 32, 0, stream>>>(agg, Wm, bm, hbuf);

    // Layer 3: agg = scatter(h2); logits = agg @ W2 + b2 -> d_out
    zero_kernel<<<zgrid, 256, 0, stream>>>((float4*)agg, n4);
    aggregate_kernel<<<agrid, 256, 0, stream>>>(hbuf, src, dst, agg);
    gemm_wmma_kernel<CLASSES, false><<<ggrid, 3 * 32, 0, stream>>>(agg, W2, b2, out);

    // log_softmax in place on d_out
    log_softmax_kernel<<<sgrid, 256, 0, stream>>>(out);
}